// GraphDecoder_26860725469214
// MI455X (gfx1250) — compile-verified
//
#include <hip/hip_runtime.h>

#define N_NODES 50000
#define N_EDGES 800000

typedef float v2f __attribute__((ext_vector_type(2)));
typedef float v8f __attribute__((ext_vector_type(8)));

// ---------------------------------------------------------------- utilities
__global__ void zero_f32(float* __restrict__ p, long long n) {
  long long i = (long long)blockIdx.x * blockDim.x + threadIdx.x;
  long long stride = (long long)gridDim.x * blockDim.x;
  for (; i < n; i += stride) p[i] = 0.0f;
}

__global__ void degree_kernel(const int* __restrict__ dst, float* __restrict__ deg) {
  int e = blockIdx.x * blockDim.x + threadIdx.x;
  if (e < N_EDGES) unsafeAtomicAdd(&deg[dst[e]], 1.0f);
}

__global__ void deg_inv_kernel(float* __restrict__ deg) {
  int i = blockIdx.x * blockDim.x + threadIdx.x;
  if (i < N_NODES) deg[i] = 1.0f / fmaxf(deg[i], 1.0f);
}

// ---------------------------------------------------- edge gather + scatter
// One thread per (edge, 4-float feature chunk): coalesced float4 gather +
// 4 non-returning GLOBAL_ATOMIC_ADD_F32 on contiguous addresses. Feature
// tables (<= 51MB) are L2-resident on MI455X (192MB L2).
template <int D_IN>
__global__ void scatter_kernel(const float* __restrict__ x,
                               const int* __restrict__ src,
                               const int* __restrict__ dst,
                               float* __restrict__ agg) {
  constexpr int CH = D_IN / 4;
  long long tid = (long long)blockIdx.x * blockDim.x + threadIdx.x;
  if (tid >= (long long)N_EDGES * CH) return;
  int e = (int)(tid / CH);
  int c = (int)(tid % CH);
  int s = src[e];
  int d = dst[e];
  const float4 v = *(const float4*)(x + (long long)s * D_IN + c * 4);
  float* p = agg + (long long)d * D_IN + c * 4;
  unsafeAtomicAdd(p + 0, v.x);
  unsafeAtomicAdd(p + 1, v.y);
  unsafeAtomicAdd(p + 2, v.z);
  unsafeAtomicAdd(p + 3, v.w);
}

// ------------------------------------------------------------- WMMA GEMM
// out[m,n] = relu( (agg[m,:]*deg_inv[m]) @ Wl + x[m,:] @ Wr + b )
// Block: 256 threads (8 waves), tile 128(M) x 64(N); wave w owns rows
// [16w,16w+16) x four 16-wide N tiles. K = [agg ; x] flattened into one
// double-buffered chunk loop (16 K per chunk, 16 WMMAs per chunk per wave).
template <int D_IN, int D_OUT, bool RELU>
__launch_bounds__(256)
__global__ void sage_gemm(const float* __restrict__ agg,
                          const float* __restrict__ x,
                          const float* __restrict__ Wl,
                          const float* __restrict__ Wr,
                          const float* __restrict__ bias,
                          const float* __restrict__ deg_inv,
                          float* __restrict__ out) {
  constexpr int AS = 20;   // A stride: halves hit disjoint bank residues (mod 4 split)
  constexpr int PS = 160;  // packed-B pair-row stride: 160 mod 64 = 32 -> halves disjoint
  constexpr int HC = D_IN / 16;  // chunks per pass
  constexpr int NC = 2 * HC;     // total chunks (agg pass + x pass)

  alignas(16) __shared__ float As[2][128 * AS];
  alignas(16) __shared__ float BsP[2][8 * PS];  // BsP[kp][col*2+{0,1}] = B[2kp|2kp+1][col]

  const int tid = threadIdx.x;
  const int lane = tid & 31;
  const int wave = tid >> 5;
  const int half = lane >> 4;
  const int lane16 = lane & 15;

  const int mBlock = blockIdx.x * 128;
  const int nBase  = blockIdx.y * 64;

  // staging register file
  const int r0  = tid >> 2;          // 0..63  (A rows, +64 for second)
  const int ac4 = (tid & 3) * 4;     // A col within 16-wide chunk
  const int krow = tid >> 4;         // 0..15  (B row)
  const int bc4 = (tid & 15) * 4;    // B col
  float4 aReg[2];
  float  sReg[2];
  float4 bReg;

  auto load_tile = [&](int chunk) {
    const bool p0 = chunk < HC;
    const float* Ag = p0 ? agg : x;
    const float* W  = p0 ? Wl  : Wr;
    const int k0 = (p0 ? chunk : chunk - HC) * 16;
#pragma unroll
    for (int rr = 0; rr < 2; ++rr) {
      const int row = mBlock + r0 + rr * 64;
      aReg[rr] = make_float4(0.f, 0.f, 0.f, 0.f);
      sReg[rr] = 1.0f;
      if (row < N_NODES) {
        aReg[rr] = *(const float4*)(Ag + (long long)row * D_IN + k0 + ac4);
        if (p0) sReg[rr] = deg_inv[row];
      }
    }
    bReg = *(const float4*)(W + (long long)(k0 + krow) * D_OUT + nBase + bc4);
  };

  auto store_tile = [&](int buf) {
#pragma unroll
    for (int rr = 0; rr < 2; ++rr) {
      float* ap = &As[buf][(r0 + rr * 64) * AS + ac4];
      ap[0] = aReg[rr].x * sReg[rr];
      ap[1] = aReg[rr].y * sReg[rr];
      ap[2] = aReg[rr].z * sReg[rr];
      ap[3] = aReg[rr].w * sReg[rr];
    }
    const int kp = krow >> 1, odd = krow & 1;
    float* bp = &BsP[buf][kp * PS + odd];
    bp[(bc4 + 0) * 2] = bReg.x;
    bp[(bc4 + 1) * 2] = bReg.y;
    bp[(bc4 + 2) * 2] = bReg.z;
    bp[(bc4 + 3) * 2] = bReg.w;
  };

  v8f acc[4];
#pragma unroll
  for (int i = 0; i < 4; ++i) acc[i] = (v8f){0.f, 0.f, 0.f, 0.f, 0.f, 0.f, 0.f, 0.f};

  auto compute_tile = [&](int buf) {
#pragma unroll
    for (int kk = 0; kk < 4; ++kk) {
      const int kb = kk * 4 + half * 2;  // A 16x4 f32 frag: halves carry K {0,1}/{2,3}
      v2f a;
      const int ai = (wave * 16 + lane16) * AS + kb;
      a.x = As[buf][ai];
      a.y = As[buf][ai + 1];
      const int bo = (kk * 2 + half) * PS + lane16 * 2;
#pragma unroll
      for (int nt = 0; nt < 4; ++nt) {
        v2f b = *(const v2f*)&BsP[buf][bo + nt * 32];  // single ds_load_b64
        acc[nt] = __builtin_amdgcn_wmma_f32_16x16x4_f32(
            false, a, false, b, (short)0, acc[nt], false, false);
      }
    }
  };

  // software pipeline: global loads for chunk i+1 in flight under compute(i)
  load_tile(0);
  store_tile(0);
  __syncthreads();
#pragma unroll 1
  for (int chunk = 0; chunk < NC; ++chunk) {
    if (chunk + 1 < NC) load_tile(chunk + 1);
    compute_tile(chunk & 1);
    __syncthreads();
    if (chunk + 1 < NC) {
      store_tile((chunk + 1) & 1);
      __syncthreads();
    }
  }

  // epilogue: D layout -> VGPR r: lanes 0-15 = (M=r, N=lane), lanes 16-31 = (M=r+8)
  const int mWave = mBlock + wave * 16;
#pragma unroll
  for (int nt = 0; nt < 4; ++nt) {
    const int col = nBase + nt * 16 + lane16;
    const float bv = bias[col];
#pragma unroll
    for (int i = 0; i < 8; ++i) {
      const int row = mWave + i + half * 8;
      if (row < N_NODES) {
        float v = acc[nt][i] + bv;
        if (RELU) v = fmaxf(v, 0.0f);
        out[(long long)row * D_OUT + col] = v;
      }
    }
  }
}

// ---------------------------------------------------------------- launcher
extern "C" void kernel_launch(void* const* d_in, const int* in_sizes, int n_in,
                              void* d_out, int out_size, void* d_ws, size_t ws_size,
                              hipStream_t stream) {
  (void)in_sizes; (void)n_in; (void)out_size; (void)ws_size;

  const float* z  = (const float*)d_in[0];
  const int* ei   = (const int*)d_in[1];
  const int* src  = ei;
  const int* dst  = ei + N_EDGES;
  const float* Wl1 = (const float*)d_in[2],  *Wr1 = (const float*)d_in[3],  *b1 = (const float*)d_in[4];
  const float* Wl2 = (const float*)d_in[5],  *Wr2 = (const float*)d_in[6],  *b2 = (const float*)d_in[7];
  const float* Wl3 = (const float*)d_in[8],  *Wr3 = (const float*)d_in[9],  *b3 = (const float*)d_in[10];
  const float* Wl4 = (const float*)d_in[11], *Wr4 = (const float*)d_in[12], *b4 = (const float*)d_in[13];
  float* out = (float*)d_out;

  // workspace: deg_inv | agg(50000x256) | hA(50000x256) | hB(50000x256)
  float* deg = (float*)d_ws;
  float* agg = deg + 50048;  // keep 16B alignment
  float* hA  = agg + (long long)N_NODES * 256;
  float* hB  = hA  + (long long)N_NODES * 256;

  const int mBlocks = (N_NODES + 127) / 128;  // 391

  // degrees
  zero_f32<<<256, 256, 0, stream>>>(deg, N_NODES);
  degree_kernel<<<(N_EDGES + 255) / 256, 256, 0, stream>>>(dst, deg);
  deg_inv_kernel<<<(N_NODES + 255) / 256, 256, 0, stream>>>(deg);

  // layer 1: 128 -> 256, relu
  zero_f32<<<2048, 256, 0, stream>>>(agg, (long long)N_NODES * 128);
  {
    long long tot = (long long)N_EDGES * (128 / 4);
    scatter_kernel<128><<<(int)((tot + 255) / 256), 256, 0, stream>>>(z, src, dst, agg);
  }
  sage_gemm<128, 256, true><<<dim3(mBlocks, 256 / 64), 256, 0, stream>>>(agg, z, Wl1, Wr1, b1, deg, hA);

  // layer 2: 256 -> 256, relu
  zero_f32<<<2048, 256, 0, stream>>>(agg, (long long)N_NODES * 256);
  {
    long long tot = (long long)N_EDGES * (256 / 4);
    scatter_kernel<256><<<(int)((tot + 255) / 256), 256, 0, stream>>>(hA, src, dst, agg);
  }
  sage_gemm<256, 256, true><<<dim3(mBlocks, 256 / 64), 256, 0, stream>>>(agg, hA, Wl2, Wr2, b2, deg, hB);

  // layer 3: 256 -> 128, relu
  zero_f32<<<2048, 256, 0, stream>>>(agg, (long long)N_NODES * 256);
  {
    long long tot = (long long)N_EDGES * (256 / 4);
    scatter_kernel<256><<<(int)((tot + 255) / 256), 256, 0, stream>>>(hB, src, dst, agg);
  }
  sage_gemm<256, 128, true><<<dim3(mBlocks, 128 / 64), 256, 0, stream>>>(agg, hB, Wl3, Wr3, b3, deg, hA);

  // layer 4: 128 -> 64, no relu
  zero_f32<<<2048, 256, 0, stream>>>(agg, (long long)N_NODES * 128);
  {
    long long tot = (long long)N_EDGES * (128 / 4);
    scatter_kernel<128><<<(int)((tot + 255) / 256), 256, 0, stream>>>(hA, src, dst, agg);
  }
  sage_gemm<128, 64, false><<<dim3(mBlocks, 64 / 64), 256, 0, stream>>>(agg, hA, Wl4, Wr4, b4, deg, out);
}